// GConv_44822278701654
// MI455X (gfx1250) — compile-verified
//
#include <hip/hip_runtime.h>

typedef __attribute__((ext_vector_type(2))) float v2f;
typedef __attribute__((ext_vector_type(8))) float v8f;

// ---------------------------------------------------------------------------
// Elementwise helpers
// ---------------------------------------------------------------------------
__global__ void fill_kernel(float* __restrict__ p, float val, int n) {
    int i = blockIdx.x * blockDim.x + threadIdx.x;
    if (i < n) p[i] = val;
}

__global__ void deg_kernel(const int* __restrict__ dst, float* __restrict__ deg, int E) {
    int e = blockIdx.x * blockDim.x + threadIdx.x;
    if (e < E) atomicAdd(&deg[dst[e]], 1.0f);   // global_atomic_add_f32, no return
}

__global__ void dis_kernel(const float* __restrict__ deg, float* __restrict__ dis, int n) {
    int i = blockIdx.x * blockDim.x + threadIdx.x;
    if (i < n) dis[i] = rsqrtf(deg[i]);         // v_rsq_f32
}

__global__ void norm_kernel(const int* __restrict__ src, const int* __restrict__ dst,
                            const float* __restrict__ dis, float* __restrict__ nrm, int E) {
    int e = blockIdx.x * blockDim.x + threadIdx.x;
    if (e < E) nrm[e] = dis[src[e]] * dis[dst[e]];
}

// ---------------------------------------------------------------------------
// fp32 GEMM via V_WMMA_F32_16X16X4_F32.
// One wave computes one 16x16 output tile; K consumed 4 at a time.
// Requires M % 16 == 0, Ncols % 16 == 0 (true here: 50000, 128, 64).
// All exits are wave-uniform, so EXEC is all-1s at every WMMA.
//
// ISA fragment layouts (CDNA5 ISA):
//   A 16x4 : lane l: m = l&15, h = l>>4;  a = { A[m][2h], A[m][2h+1] }
//   B 4x16 : lane l: n = l&15, h = l>>4;  b = { B[2h][n], B[2h+1][n] }
//   C 16x16: VGPR v, lanes 0-15 -> row v, lanes 16-31 -> row v+8, col = l&15
// ---------------------------------------------------------------------------
__global__ __launch_bounds__(256) void gemm_wmma_f32(
    const float* __restrict__ A, const float* __restrict__ B, float* __restrict__ C,
    int M, int K, int Ncols) {
    int gwave = (blockIdx.x * blockDim.x + threadIdx.x) >> 5;
    int lane  = threadIdx.x & 31;
    int tilesN = Ncols >> 4;
    int totalTiles = (M >> 4) * tilesN;
    if (gwave >= totalTiles) return;            // wave-uniform guard

    int tileM = gwave / tilesN;
    int tileN = gwave - tileM * tilesN;

    int m = lane & 15;
    int h = lane >> 4;                          // 0: K pair {0,1}, 1: K pair {2,3}

    const float* Arow = A + (size_t)(tileM * 16 + m) * K;
    const float* Bcol = B + tileN * 16 + m;     // (lane&15) doubles as column index

    v8f acc = {};
    #pragma unroll 4
    for (int kb = 0; kb < K; kb += 4) {
        int k0 = kb + 2 * h;
        v2f a, b;
        a.x = Arow[k0];
        a.y = Arow[k0 + 1];
        b.x = Bcol[(size_t)k0 * Ncols];
        b.y = Bcol[(size_t)(k0 + 1) * Ncols];
        acc = __builtin_amdgcn_wmma_f32_16x16x4_f32(
            /*neg_a=*/false, a, /*neg_b=*/false, b,
            /*c_mod=*/(short)0, acc, /*reuse_a=*/false, /*reuse_b=*/false);
    }

    float* Cp = C + (size_t)(tileM * 16) * Ncols + tileN * 16 + m;
    #pragma unroll
    for (int v = 0; v < 8; ++v)
        Cp[(size_t)(v + 8 * h) * Ncols] = acc[v];
}

// ---------------------------------------------------------------------------
// Edge scatter: agg[dst] += xw[src] * norm[e].
// CHUNKS = D/4 lanes per edge, float4 gather + 4 float atomics per lane.
// ---------------------------------------------------------------------------
template <int LOG_CHUNKS>
__global__ void scatter_kernel(const int* __restrict__ src, const int* __restrict__ dst,
                               const float* __restrict__ nrm, const float* __restrict__ xw,
                               float* __restrict__ agg, int E) {
    constexpr int CHUNKS = 1 << LOG_CHUNKS;     // D = 4 * CHUNKS
    constexpr int D = CHUNKS * 4;
    int t = blockIdx.x * blockDim.x + threadIdx.x;
    int e = t >> LOG_CHUNKS;
    int c = t & (CHUNKS - 1);
    if (e >= E) return;
    int s = src[e], d = dst[e];
    float w = nrm[e];
    float4 v = reinterpret_cast<const float4*>(xw + (size_t)s * D)[c];
    float* o = agg + (size_t)d * D + c * 4;
    atomicAdd(o + 0, v.x * w);
    atomicAdd(o + 1, v.y * w);
    atomicAdd(o + 2, v.z * w);
    atomicAdd(o + 3, v.w * w);
}

// z = relu(agg + xw * dis^2 + bias), written over agg
__global__ void self_bias_relu_kernel(float* __restrict__ agg, const float* __restrict__ xw,
                                      const float* __restrict__ dis, const float* __restrict__ bias,
                                      int total, int logD) {
    int i = blockIdx.x * blockDim.x + threadIdx.x;
    if (i >= total) return;
    int node = i >> logD;
    int f = i & ((1 << logD) - 1);
    float dd = dis[node];
    float v = agg[i] + xw[i] * dd * dd + bias[f];
    agg[i] = v > 0.0f ? v : 0.0f;
}

// out = agg + xw * dis^2 + bias (no relu, final layer)
__global__ void self_bias_kernel(float* __restrict__ agg, const float* __restrict__ xw,
                                 const float* __restrict__ dis, const float* __restrict__ bias,
                                 int total, int logD) {
    int i = blockIdx.x * blockDim.x + threadIdx.x;
    if (i >= total) return;
    int node = i >> logD;
    int f = i & ((1 << logD) - 1);
    float dd = dis[node];
    agg[i] = agg[i] + xw[i] * dd * dd + bias[f];
}

// ---------------------------------------------------------------------------
extern "C" void kernel_launch(void* const* d_in, const int* in_sizes, int n_in,
                              void* d_out, int out_size, void* d_ws, size_t ws_size,
                              hipStream_t stream) {
    const float* x  = (const float*)d_in[0];
    const int*   ei = (const int*)d_in[1];
    const float* W1 = (const float*)d_in[2];
    const float* b1 = (const float*)d_in[3];
    const float* W2 = (const float*)d_in[4];
    const float* b2 = (const float*)d_in[5];

    const int D_IN = 128, D_HID = 128, D_OUT = 64;
    int N = in_sizes[0] / D_IN;
    int E = in_sizes[1] / 2;
    const int* src = ei;        // edge_index[0]
    const int* dst = ei + E;    // edge_index[1]

    // Workspace layout (all row bases 16B aligned: 2N+E = 900000 floats here)
    float* ws   = (float*)d_ws;
    float* deg  = ws;  ws += N;
    float* dis  = ws;  ws += N;
    float* nrm  = ws;  ws += E;
    float* xw1  = ws;  ws += (size_t)N * D_HID;
    float* agg1 = ws;  ws += (size_t)N * D_HID;   // becomes z after relu
    float* xw2  = ws;  ws += (size_t)N * D_OUT;
    float* out  = (float*)d_out;

    const int T = 256;
    auto blocks = [](long n, int t) { return (int)((n + t - 1) / t); };

    // ---- normalization coefficients ----
    fill_kernel<<<blocks(N, T), T, 0, stream>>>(deg, 1.0f, N);        // self-loop
    deg_kernel<<<blocks(E, T), T, 0, stream>>>(dst, deg, E);
    dis_kernel<<<blocks(N, T), T, 0, stream>>>(deg, dis, N);
    norm_kernel<<<blocks(E, T), T, 0, stream>>>(src, dst, dis, nrm, E);

    // ---- layer 1: xw1 = X @ W1 (WMMA fp32) ----
    {
        long waves = (long)(N / 16) * (D_HID / 16);
        gemm_wmma_f32<<<blocks(waves * 32, T), T, 0, stream>>>(x, W1, xw1, N, D_IN, D_HID);
    }
    fill_kernel<<<blocks((long)N * D_HID, T), T, 0, stream>>>(agg1, 0.0f, N * D_HID);
    scatter_kernel<5><<<blocks((long)E * 32, T), T, 0, stream>>>(src, dst, nrm, xw1, agg1, E);
    self_bias_relu_kernel<<<blocks((long)N * D_HID, T), T, 0, stream>>>(
        agg1, xw1, dis, b1, N * D_HID, 7);

    // ---- layer 2: xw2 = z @ W2 (WMMA fp32) ----
    {
        long waves = (long)(N / 16) * (D_OUT / 16);
        gemm_wmma_f32<<<blocks(waves * 32, T), T, 0, stream>>>(agg1, W2, xw2, N, D_HID, D_OUT);
    }
    fill_kernel<<<blocks((long)N * D_OUT, T), T, 0, stream>>>(out, 0.0f, N * D_OUT);
    scatter_kernel<4><<<blocks((long)E * 16, T), T, 0, stream>>>(src, dst, nrm, xw2, out, E);
    self_bias_kernel<<<blocks((long)N * D_OUT, T), T, 0, stream>>>(
        out, xw2, dis, b2, N * D_OUT, 6);
}